// Canny_1116691497316
// MI455X (gfx1250) — compile-verified
//
#include <hip/hip_runtime.h>
#include <stdint.h>

// Canny pipeline for MI455X (gfx1250, wave32).
//   K1: fused blur+sobel+mag2+NMS+threshold, 32x32 tiles.
//       - raw tile staged with CDNA5 async global->LDS (ASYNCcnt), overlapped
//         with building the banded gaussian weight matrix Wv in LDS
//       - vertical gaussian pass on the matrix pipe: V_WMMA_F32_16X16X4_F32,
//         branch-free k-loop (ds_load_2addr + wmma) and unconditional D scatter
//         into a padded LDS tile (no EXEC divergence around matrix ops)
//       - direction quantization via slope compares (no atan2), mag^2 (no sqrt)
//   K2: hysteresis relaxation (monotone morphological reconstruction), 16 passes
//   K3: out = x[c] * 0.5 * code
// Workspace: edge codes uint8[32*512*512] at d_ws, fully rewritten each call.

#define IMG_H 512
#define IMG_W 512
#define IMG_HW (IMG_H * IMG_W)
#define NSEL 32

// 5-tap gaussian, sigma=1, normalized: exp(-0.5*t^2)/sum
#define G0 0.05448868454f
#define G1 0.24420134160f
#define G2 0.40261994772f

typedef __attribute__((ext_vector_type(2))) float v2f;
typedef __attribute__((ext_vector_type(8))) float v8f;

__device__ __forceinline__ int reflect_idx(int i) {
  // jnp.pad mode='reflect' (no edge repeat): -1->1, -2->2, H->H-2, H+1->H-3
  i = (i < 0) ? -i : i;
  i = (i >= IMG_H) ? (2 * IMG_H - 2 - i) : i;
  return i;
}

__device__ __forceinline__ unsigned lds_lo32(const void* p) {
  // generic address of a __shared__ object: low 32 bits == wave-relative LDS byte address
  return (unsigned)(unsigned long long)(uintptr_t)p;
}

__device__ __forceinline__ void async_load_f32(unsigned lds_byte_addr, const float* gaddr) {
  // CDNA5 async copy global -> LDS (per-lane gather), tracked with ASYNCcnt
  asm volatile("global_load_async_to_lds_b32 %0, %1, off"
               :: "v"(lds_byte_addr), "v"(gaddr) : "memory");
}

__device__ __forceinline__ void wait_async0() {
  asm volatile("s_wait_asynccnt 0" ::: "memory");
}

// ---------------------------------------------------------------------------
// K1: fused front end. grid (16,16,32), block 256 (8 wave32 waves).
// ---------------------------------------------------------------------------
__global__ __launch_bounds__(256) void canny_front(
    const float* __restrict__ x, const long long* __restrict__ indices,
    unsigned char* __restrict__ edges)
{
  const int tx0 = blockIdx.x * 32;
  const int ty0 = blockIdx.y * 32;
  const int k   = blockIdx.z;
  const int c   = (int)indices[k];
  const float* img = x + (size_t)c * IMG_HW;
  const int tid = threadIdx.x;

  // strides 41/37/49/35 are co-prime with 64 LDS banks
  __shared__ float s_raw[40][41];   // raw, reflect-indexed, region [t0-4, t0+35]
  __shared__ float s_wv[48][41];    // banded gaussian weights, rows 36..47 zero
  __shared__ float s_tmp[40][37];   // horizontal blur, 40 rows x 36 cols (cols clamped)
  __shared__ float s_blur[48][49];  // blur at clamped coords; rows/cols >=36 are WMMA pad
  __shared__ float s_mag[34][35];   // squared grad magnitude (+eps; 0 outside image)

  // ---- stage A: async gather of reflect-padded raw tile into LDS
  for (int i = tid; i < 40 * 40; i += 256) {
    int yl = i / 40, xl = i - (i / 40) * 40;
    int ry = reflect_idx(ty0 - 4 + yl);
    int rx = reflect_idx(tx0 - 4 + xl);
    async_load_f32(lds_lo32(&s_raw[yl][xl]), img + ry * IMG_W + rx);
  }

  // ---- overlap: build Wv(48x40) while the async copies are in flight.
  // Wv[row][kg] = gaussian tap for Blur(row) reading Tmp(kg); rows clamped at
  // image borders (sobel replicate-pad halo), pad rows 36..47 are zero.
  for (int i = tid; i < 48 * 40; i += 256) {
    int row = i / 40, kg = i - (i / 40) * 40;
    int ey = ty0 - 2 + row;
    ey = ey < 0 ? 0 : (ey > IMG_H - 1 ? IMG_H - 1 : ey);
    int d = kg - (ey - (ty0 - 4));
    int ad = d < 0 ? -d : d;
    float w = (ad == 0) ? G2 : ((ad == 1) ? G1 : ((ad == 2) ? G0 : 0.0f));
    s_wv[row][kg] = (row < 36) ? w : 0.0f;
  }

  wait_async0();
  __syncthreads();

  // ---- stage B: horizontal 5-tap blur on VALU (reflect baked into s_raw layout)
  for (int i = tid; i < 40 * 36; i += 256) {
    int r = i / 36, xl = i - (i / 36) * 36;
    int ex = tx0 - 2 + xl;                       // virtual column, clamp for sobel halo
    ex = ex < 0 ? 0 : (ex > IMG_W - 1 ? IMG_W - 1 : ex);
    int b = ex - (tx0 - 4);
    s_tmp[r][xl] = G0 * (s_raw[r][b - 2] + s_raw[r][b + 2])
                 + G1 * (s_raw[r][b - 1] + s_raw[r][b + 1])
                 + G2 *  s_raw[r][b];
  }
  __syncthreads();

  // ---- stage C: vertical 5-tap blur on the matrix pipe.
  // Blur(36x36) = Wv(36x40) x Tmp(40x36): 3x3 tiles of 16x16 output, K=40 as
  // 10 steps of V_WMMA_F32_16X16X4_F32, distributed over the 8 waves.
  // A layout: lane l holds A[M=l&15][K=(l>>4)*2 + vgpr]
  // B layout: lane l holds B[K=(l>>4)*2 + vgpr][N=l&15]
  {
    const int wave = tid >> 5;           // wave-uniform
    const int lane = tid & 31;
    const int lrow = lane & 15;          // M index for A, N index for B
    const int khalf = (lane >> 4) * 2;   // K base held by this lane half
    for (int t = wave; t < 9; t += 8) {  // wave-uniform loop, EXEC stays all-1s
      const int mt = t / 3, nt = t - (t / 3) * 3;
      const int row = mt * 16 + lrow;    // A row (pad rows hold zero weights)
      int col = nt * 16 + lrow;          // B column this lane supplies
      col = col < 36 ? col : 35;         // keep LDS reads in-bounds (pad cols dup col 35)
      v8f acc = {};
#pragma unroll
      for (int kk = 0; kk < 10; ++kk) {
        const int kg0 = kk * 4 + khalf;  // this lane supplies K = kg0, kg0+1
        v2f a = {s_wv[row][kg0], s_wv[row][kg0 + 1]};
        v2f b = {s_tmp[kg0][col], s_tmp[kg0 + 1][col]};
        acc = __builtin_amdgcn_wmma_f32_16x16x4_f32(
            false, a, false, b, (short)0, acc, false, false);
      }
      // D layout: VGPR v, lane l -> M = v + (l>>4)*8, N = l&15.
      // s_blur is padded to 48x49, so the scatter is unconditional: pad rows
      // get zeros (zero Wv rows), pad cols duplicates of col 35; never read.
#pragma unroll
      for (int v = 0; v < 8; ++v) {
        s_blur[mt * 16 + v + (lane >> 4) * 8][nt * 16 + lrow] = acc[v];
      }
    }
  }
  __syncthreads();

  // ---- stage D: sobel + squared magnitude on 34x34 (0 outside image: NMS zero-pad)
  for (int i = tid; i < 34 * 34; i += 256) {
    int ym = i / 34, xm = i - (i / 34) * 34;
    int yi = ty0 - 1 + ym;
    int xi = tx0 - 1 + xm;
    float m2 = 0.0f;
    if (yi >= 0 && yi < IMG_H && xi >= 0 && xi < IMG_W) {
      int by = ym + 1, bx = xm + 1;
      float tl = s_blur[by - 1][bx - 1], tc = s_blur[by - 1][bx], tr = s_blur[by - 1][bx + 1];
      float ml = s_blur[by    ][bx - 1],                          mr = s_blur[by    ][bx + 1];
      float bl = s_blur[by + 1][bx - 1], bc = s_blur[by + 1][bx], br = s_blur[by + 1][bx + 1];
      float gxv = (tr + 2.0f * mr + br) - (tl + 2.0f * ml + bl);
      float gyv = (bl + 2.0f * bc + br) - (tl + 2.0f * tc + tr);
      m2 = gxv * gxv + gyv * gyv + 1e-6f;   // sqrt is monotone: keep mag^2 everywhere
    }
    s_mag[ym][xm] = m2;
  }
  __syncthreads();

  // ---- stage E: NMS + double threshold -> code {0,1,2}
  unsigned char* erow = edges + (size_t)k * IMG_HW;
  for (int i = tid; i < 32 * 32; i += 256) {
    int oy = i >> 5, ox = i & 31;
    int by = oy + 2, bx = ox + 2;
    float tl = s_blur[by - 1][bx - 1], tc = s_blur[by - 1][bx], tr = s_blur[by - 1][bx + 1];
    float ml = s_blur[by    ][bx - 1],                          mr = s_blur[by    ][bx + 1];
    float bl = s_blur[by + 1][bx - 1], bc = s_blur[by + 1][bx], br = s_blur[by + 1][bx + 1];
    float gxv = (tr + 2.0f * mr + br) - (tl + 2.0f * ml + bl);
    float gyv = (bl + 2.0f * bc + br) - (tl + 2.0f * tc + tr);

    // round(atan2(gy,gx)*4/pi) mod 4 selects the NMS axis; min(sp,sn) is
    // symmetric under direction negation, so only the axis matters.
    // Sector boundaries are odd multiples of pi/8 -> slope compares.
    float ax = fabsf(gxv), ay = fabsf(gyv);
    int dy, dx;
    if (ay <= 0.41421356237309515f * ax)      { dy = 0; dx = 1; }   // E-W
    else if (ay >= 2.4142135623730951f * ax)  { dy = 1; dx = 0; }   // N-S
    else if ((gxv >= 0.0f) == (gyv >= 0.0f))  { dy = 1; dx = 1; }   // SE-NW diagonal
    else                                      { dy = 1; dx = -1; }  // SW-NE diagonal

    float m2 = s_mag[oy + 1][ox + 1];
    float sp = m2 - s_mag[oy + 1 + dy][ox + 1 + dx];
    float sn = m2 - s_mag[oy + 1 - dy][ox + 1 - dx];
    bool keep = (sp > 0.0f) && (sn > 0.0f);

    // thresholds on mag: 0.1 / 0.2  ->  on mag^2: 0.01 / 0.04
    unsigned char code = 0;
    if (keep && m2 > 0.01f) code = (m2 > 0.04f) ? (unsigned char)2 : (unsigned char)1;
    erow[(ty0 + oy) * IMG_W + (tx0 + ox)] = code;
  }
}

// ---------------------------------------------------------------------------
// K2: hysteresis relaxation pass. grid (8,8,32), block 256, 64x64 tile + halo.
// Weak(1)->Strong(2) promotion is monotone, so in-place chaotic iteration
// converges to the reference fixed point; intra-tile convergence per pass,
// fixed number of passes for inter-tile propagation (graph-capture safe).
// ---------------------------------------------------------------------------
__global__ __launch_bounds__(256) void hyst_relax(unsigned char* __restrict__ edges)
{
  const int tx0 = blockIdx.x * 64;
  const int ty0 = blockIdx.y * 64;
  const int k = blockIdx.z;
  unsigned char* e = edges + (size_t)k * IMG_HW;
  const int tid = threadIdx.x;

  __shared__ unsigned char s_e[66][68];
  __shared__ int s_changed;

  for (int i = tid; i < 66 * 66; i += 256) {
    int yl = i / 66, xl = i - (i / 66) * 66;
    int y = ty0 - 1 + yl, x = tx0 - 1 + xl;
    unsigned char v = 0;
    if (y >= 0 && y < IMG_H && x >= 0 && x < IMG_W) v = e[y * IMG_W + x];
    s_e[yl][xl] = v;
  }
  __syncthreads();

  for (int it = 0; it < 96; ++it) {
    if (tid == 0) s_changed = 0;
    __syncthreads();
    int ch = 0;
#pragma unroll 4
    for (int j = 0; j < 16; ++j) {
      int p = tid + j * 256;
      int y = p >> 6, x = p & 63;
      if (s_e[y + 1][x + 1] == (unsigned char)1) {
        int strong =
            (s_e[y][x] == 2)     | (s_e[y][x + 1] == 2)     | (s_e[y][x + 2] == 2) |
            (s_e[y + 1][x] == 2) |                            (s_e[y + 1][x + 2] == 2) |
            (s_e[y + 2][x] == 2) | (s_e[y + 2][x + 1] == 2) | (s_e[y + 2][x + 2] == 2);
        if (strong) { s_e[y + 1][x + 1] = (unsigned char)2; ch = 1; }
      }
    }
    if (ch) s_changed = 1;
    __syncthreads();
    if (s_changed == 0) break;
  }

  for (int j = 0; j < 16; ++j) {
    int p = tid + j * 256;
    int y = p >> 6, x = p & 63;
    e[(ty0 + y) * IMG_W + (tx0 + x)] = s_e[y + 1][x + 1];
  }
}

// ---------------------------------------------------------------------------
// K3: out[k] = x[indices[k]] * 0.5 * code, float4/uchar4 vectorized.
// ---------------------------------------------------------------------------
__global__ __launch_bounds__(256) void canny_finalize(
    const float* __restrict__ x, const long long* __restrict__ indices,
    const unsigned char* __restrict__ edges, float* __restrict__ out)
{
  const int per_img = IMG_HW / 4;
  int i = blockIdx.x * 256 + threadIdx.x;   // float4 index
  int k = i / per_img;
  if (k >= NSEL) return;
  int p = i - k * per_img;
  int c = (int)indices[k];
  float4 xv = reinterpret_cast<const float4*>(x + (size_t)c * IMG_HW)[p];
  uchar4 ev = reinterpret_cast<const uchar4*>(edges + (size_t)k * IMG_HW)[p];
  float4 o;
  o.x = xv.x * (0.5f * (float)ev.x);
  o.y = xv.y * (0.5f * (float)ev.y);
  o.z = xv.z * (0.5f * (float)ev.z);
  o.w = xv.w * (0.5f * (float)ev.w);
  reinterpret_cast<float4*>(out + (size_t)k * IMG_HW)[p] = o;
}

extern "C" void kernel_launch(void* const* d_in, const int* in_sizes, int n_in,
                              void* d_out, int out_size, void* d_ws, size_t ws_size,
                              hipStream_t stream) {
  (void)in_sizes; (void)n_in; (void)out_size; (void)ws_size;
  const float* x = (const float*)d_in[0];
  // d_in[1] = params (int64, always 1 in harness; params==0 branch is shape-
  // incompatible with out_size, so only the main path is implemented)
  const long long* indices = (const long long*)d_in[2];
  unsigned char* edges = (unsigned char*)d_ws;   // 32*512*512 bytes
  float* out = (float*)d_out;

  dim3 blk(256);
  canny_front<<<dim3(IMG_W / 32, IMG_H / 32, NSEL), blk, 0, stream>>>(x, indices, edges);
  for (int pass = 0; pass < 16; ++pass) {
    hyst_relax<<<dim3(IMG_W / 64, IMG_H / 64, NSEL), blk, 0, stream>>>(edges);
  }
  canny_finalize<<<dim3((NSEL * (IMG_HW / 4) + 255) / 256), blk, 0, stream>>>(x, indices, edges, out);
}